// get_model_84610855731463
// MI455X (gfx1250) — compile-verified
//
#include <hip/hip_runtime.h>
#include <math.h>

typedef __attribute__((ext_vector_type(2))) float v2f;
typedef __attribute__((ext_vector_type(4))) float v4f;
typedef __attribute__((ext_vector_type(8))) float v8f;

#define NODE_C 160
#define EDGE_C 27
#define KPAD   28        // EDGE_C padded to multiple of 4 for 16x16x4 WMMA
#define NPAIR  (KPAD / 2)   // 14 K-pair rows
#define NCOLS  512
#define TOPK   5

#define PAIR_STRIDE 528  // float2 per K-pair row (512 used + 16 pad):
                         // 4224B stride => halves hit disjoint bank-pair sets
#define CT_STRIDE   516  // 512 + 4 : rows j vs j+8 -> disjoint LDS bank sets

// ---------------------------------------------------------------------------
// Edge path: one-hot[16 x 28] (f32) x table[28 x 512] (f32) via
// V_WMMA_F32_16X16X4_F32 (exact f32, matches reference numerics).
// One workgroup = 16 output rows x 512 cols. 4 waves x 8 N-tiles each,
// 7 K-steps per tile. Table staged in LDS K-pair-interleaved so every
// A/B fragment is ONE aligned ds_load_b64 into a WMMA-ready VGPR pair.
// C tile staged in LDS, written back as coalesced v4f nontemporal stores.
// ---------------------------------------------------------------------------
__global__ __launch_bounds__(128) void edge_topk_wmma_kernel(
    const float* __restrict__ W,   // [E, 27]
    const float* __restrict__ T,   // [27, 512]
    float* __restrict__ out,       // [E, 512]
    int E) {
  __shared__ v2f   tblp[NPAIR][PAIR_STRIDE];   // 59136 B, {T[2p][c],T[2p+1][c]}
  __shared__ v2f   onehp[16][NPAIR];           //  1792 B, one-hot K-pairs
  __shared__ float ctile[16][CT_STRIDE];       // 33024 B

  const int tid = threadIdx.x;

  // --- stage table K-pair-interleaved; 512B contiguous per wave load, 1024B per wave store pair ---
  {
    const v4f* gt = (const v4f*)T;             // 27 rows x 128 v4f
    const v4f z4 = {0.f, 0.f, 0.f, 0.f};
#pragma unroll 1
    for (int p = 0; p < NPAIR; ++p) {
      const v4f a4 = gt[(2 * p) * (NCOLS / 4) + tid];          // rows 0..26
      const v4f b4 = (2 * p + 1 < EDGE_C)
                         ? gt[(2 * p + 1) * (NCOLS / 4) + tid] // rows 1..25
                         : z4;                                 // pad row 27
      const v4f lo = {a4.x, b4.x, a4.y, b4.y};
      const v4f hi = {a4.z, b4.z, a4.w, b4.w};
      v4f* dst = (v4f*)&tblp[p][4 * tid];      // 4 float2 = 32B per thread
      dst[0] = lo;
      dst[1] = hi;
    }
  }

  const int rowbase = blockIdx.x * 16;

  // --- per-row top-5 over 27 weights, fully register-resident ---
  if (tid < 16) {
    const int row = rowbase + tid;
    float w[EDGE_C];
#pragma unroll
    for (int i = 0; i < EDGE_C; ++i)
      w[i] = (row < E) ? W[(size_t)row * EDGE_C + i] : -INFINITY;

    int b0 = 0, b1 = 0, b2 = 0, b3 = 0, b4 = 0;
#pragma unroll
    for (int p = 0; p < TOPK; ++p) {
      float bv = -INFINITY;
      int bi = 0;
#pragma unroll
      for (int i = 0; i < EDGE_C; ++i) {
        if (w[i] > bv) { bv = w[i]; bi = i; }   // first-occurrence tie break
      }
#pragma unroll
      for (int i = 0; i < EDGE_C; ++i)
        w[i] = (i == bi) ? -INFINITY : w[i];
      if (p == 0) b0 = bi; else if (p == 1) b1 = bi; else if (p == 2) b2 = bi;
      else if (p == 3) b3 = bi; else b4 = bi;
    }

    const float live = (row < E) ? 1.0f : 0.0f; // tail rows contribute zero
#pragma unroll
    for (int p = 0; p < NPAIR; ++p) {
      const int i0 = 2 * p, i1 = 2 * p + 1;
      v2f pv;
      pv.x = ((i0 == b0) | (i0 == b1) | (i0 == b2) | (i0 == b3) | (i0 == b4))
                 ? live : 0.0f;
      pv.y = ((i1 == b0) | (i1 == b1) | (i1 == b2) | (i1 == b3) | (i1 == b4))
                 ? live : 0.0f;
      onehp[tid][p] = pv;
    }
  }
  __syncthreads();   // EXEC uniform all-1s from here through the WMMA region

  const int lane = tid & 31;
  const int wave = tid >> 5;
  const int m    = lane & 15;   // A row / B col / C col within tile
  const int half = lane >> 4;   // selects K pair (A/B) or +8 rows (C)

  // A fragments: identical across all 8 N-tiles of this wave -> hoist.
  // One ds_load_b64 each (pair index = 2*ks + half <=> K rows kb, kb+1).
  v2f a[7];
#pragma unroll
  for (int ks = 0; ks < 7; ++ks)
    a[ks] = onehp[m][2 * ks + half];

#pragma unroll
  for (int nt = 0; nt < 8; ++nt) {
    const int col = (wave * 8 + nt) * 16 + m;

    v8f acc = {};
#pragma unroll
    for (int ks = 0; ks < 7; ++ks) {
      const v2f bv = tblp[2 * ks + half][col];  // single aligned ds_load_b64
      acc = __builtin_amdgcn_wmma_f32_16x16x4_f32(
          /*neg_a=*/false, a[ks], /*neg_b=*/false, bv,
          /*c_mod=*/(short)0, acc, /*reuse_a=*/false, /*reuse_b=*/false);
    }

    // C layout: VGPR j -> tile row (j + 8*half); stage into LDS.
#pragma unroll
    for (int j = 0; j < 8; ++j)
      ctile[j + 8 * half][col] = acc[j];
  }
  __syncthreads();

  // --- coalesced writeback: one 2048B output row per iteration;
  //     each wave issues a single 512B-contiguous global_store_b128. ---
#pragma unroll 1
  for (int k = 0; k < 16; ++k) {
    const int r = rowbase + k;
    if (r < E) {                                 // uniform branch (tail only)
      const v4f v = ((const v4f*)&ctile[k][0])[tid];
      v4f* dst = (v4f*)(out + (size_t)r * NCOLS) + tid;
      __builtin_nontemporal_store(v, dst);       // streaming: don't pollute L2
    }
  }
}

// ---------------------------------------------------------------------------
// Node path: 600 rows (~0.16% of traffic). Table (327 KB) is L2-resident;
// straight gather-sum of 5 rows. One workgroup per row.
// ---------------------------------------------------------------------------
__global__ __launch_bounds__(128) void node_topk_gather_kernel(
    const float* __restrict__ W,   // [N, 160]
    const float* __restrict__ T,   // [160, 512]
    float* __restrict__ out,       // [N, 512]
    int N) {
  __shared__ float wrow[NODE_C];
  __shared__ int idx[TOPK];

  const int row = blockIdx.x;
  if (row >= N) return;

  for (int i = threadIdx.x; i < NODE_C; i += blockDim.x)
    wrow[i] = W[(size_t)row * NODE_C + i];
  __syncthreads();

  if (threadIdx.x == 0) {
    for (int p = 0; p < TOPK; ++p) {
      float bv = -INFINITY;
      int bi = 0;
      for (int i = 0; i < NODE_C; ++i) {
        float v = wrow[i];
        if (v > bv) { bv = v; bi = i; }
      }
      idx[p] = bi;
      wrow[bi] = -INFINITY;
    }
  }
  __syncthreads();

  const int i0 = idx[0], i1 = idx[1], i2 = idx[2], i3 = idx[3], i4 = idx[4];
  for (int c = threadIdx.x; c < NCOLS; c += blockDim.x) {
    float s = T[(size_t)i0 * NCOLS + c] + T[(size_t)i1 * NCOLS + c] +
              T[(size_t)i2 * NCOLS + c] + T[(size_t)i3 * NCOLS + c] +
              T[(size_t)i4 * NCOLS + c];
    out[(size_t)row * NCOLS + c] = s;
  }
}

extern "C" void kernel_launch(void* const* d_in, const int* in_sizes, int n_in,
                              void* d_out, int out_size, void* d_ws, size_t ws_size,
                              hipStream_t stream) {
  const float* node_w  = (const float*)d_in[0];  // [N, 160]
  const float* edge_w  = (const float*)d_in[1];  // [E, 27]
  const float* knode_t = (const float*)d_in[2];  // [160, 512]
  const float* kedge_t = (const float*)d_in[3];  // [27, 512]

  const int N = in_sizes[0] / NODE_C;
  const int E = in_sizes[1] / EDGE_C;

  float* out      = (float*)d_out;
  float* node_out = out;                          // [N, 512]
  float* edge_out = out + (size_t)N * NCOLS;      // [E, 512]

  node_topk_gather_kernel<<<N, 128, 0, stream>>>(node_w, knode_t, node_out, N);

  const int eblocks = (E + 15) / 16;
  edge_topk_wmma_kernel<<<eblocks, 128, 0, stream>>>(edge_w, kedge_t, edge_out, E);
}